// DiffLogicLayer_81123342287625
// MI455X (gfx1250) — compile-verified
//
#include <hip/hip_runtime.h>
#include <cstdint>
#include <cstddef>

#define IN_DIM  4096
#define OUT_DIM 8192
#define BATCH   4096
#define TB      4        // batch rows staged in LDS per block (4 * 16KB = 64KB)

typedef __attribute__((ext_vector_type(2))) float v2f;
typedef __attribute__((ext_vector_type(8))) float v8f;

// op i -> coefficients of {1, a, b, a*b} (compile-time table, never touches memory)
constexpr float kM[16][4] = {
  {0.f, 0.f, 0.f, 0.f},  // 0
  {0.f, 0.f, 0.f, 1.f},  // ab
  {0.f, 1.f, 0.f,-1.f},  // a-ab
  {0.f, 1.f, 0.f, 0.f},  // a
  {0.f, 0.f, 1.f,-1.f},  // b-ab
  {0.f, 0.f, 1.f, 0.f},  // b
  {0.f, 1.f, 1.f,-2.f},  // a+b-2ab
  {0.f, 1.f, 1.f,-1.f},  // a+b-ab
  {1.f,-1.f,-1.f, 1.f},  // 1-(a+b-ab)
  {1.f,-1.f,-1.f, 2.f},  // 1-(a+b-2ab)
  {1.f, 0.f,-1.f, 0.f},  // 1-b
  {1.f, 0.f,-1.f, 1.f},  // 1-b+ab
  {1.f,-1.f, 0.f, 0.f},  // 1-a
  {1.f,-1.f, 0.f, 1.f},  // 1-a+ab
  {1.f, 0.f, 0.f,-1.f},  // 1-ab
  {1.f, 0.f, 0.f, 0.f}   // 1
};

// B[row][n] as straight-line arithmetic over one-hot indicators: the constexpr
// coefficients (0, +-1, +-2) fold so each row costs 0-3 VALU ops. No branches.
template<int R>
__device__ __forceinline__ float brow(float i0, float i1, float i2, float i3) {
  float v = 0.f;
  if constexpr (kM[R][0] != 0.f) v = fmaf(kM[R][0], i0, v);
  if constexpr (kM[R][1] != 0.f) v = fmaf(kM[R][1], i1, v);
  if constexpr (kM[R][2] != 0.f) v = fmaf(kM[R][2], i2, v);
  if constexpr (kM[R][3] != 0.f) v = fmaf(kM[R][3], i3, v);
  return v;
}

// coef[o][0..3] = softmax(weights[o][0..15]) @ M, via V_WMMA_F32_16X16X4_F32.
// One wave32 handles 16 output rows: D(16x16) = sum_j A_j(16x4) B_j(4x16) + C,
// B columns >= 4 are zero so only D[:,0:4] is meaningful. Exact f32 math.
__global__ __launch_bounds__(256) void coef_kernel(const float* __restrict__ w,
                                                   float* __restrict__ coef) {
  const int  lane  = threadIdx.x & 31;
  const int  wave  = threadIdx.x >> 5;
  const int  obase = blockIdx.x * 128 + wave * 16;
  const int  row   = obase + (lane & 15);
  const bool hi    = (lane & 16) != 0;     // lanes 16-31 carry K=2,3 per ISA layout
  const int  n     = lane & 15;            // B/D column index

  // one-hot column indicators + half-select as floats: pure cndmask, no branch
  const float i0 = (n == 0) ? 1.f : 0.f;
  const float i1 = (n == 1) ? 1.f : 0.f;
  const float i2 = (n == 2) ? 1.f : 0.f;
  const float i3 = (n == 3) ? 1.f : 0.f;
  const float h  = hi ? 1.f : 0.f;

  // per-lane softmax numerators of row's 16 weights (lanes L and L+16 redo same row)
  const float4* wr4 = (const float4*)(w + (size_t)row * 16);
  float e[16];
  {
    const float4 w0 = wr4[0], w1 = wr4[1], w2 = wr4[2], w3 = wr4[3];
    e[0]=w0.x; e[1]=w0.y; e[2]=w0.z; e[3]=w0.w;
    e[4]=w1.x; e[5]=w1.y; e[6]=w1.z; e[7]=w1.w;
    e[8]=w2.x; e[9]=w2.y; e[10]=w2.z; e[11]=w2.w;
    e[12]=w3.x; e[13]=w3.y; e[14]=w3.z; e[15]=w3.w;
  }
  float m = e[0];
  #pragma unroll
  for (int i = 1; i < 16; ++i) m = fmaxf(m, e[i]);
  float sum = 0.f;
  #pragma unroll
  for (int i = 0; i < 16; ++i) { e[i] = expf(e[i] - m); sum += e[i]; }
  const float inv = 1.0f / sum;

  v8f acc = {0.f, 0.f, 0.f, 0.f, 0.f, 0.f, 0.f, 0.f};

  // A element: cndmask on 'hi' then scale; B element: lerp between the two
  // constexpr-folded row combos via h. Entirely branch-free straight-line code.
#define WMMA_STEP(J)                                                          \
  {                                                                           \
    v2f a, b;                                                                 \
    a.x = (hi ? e[4*(J)+2] : e[4*(J)+0]) * inv;                               \
    a.y = (hi ? e[4*(J)+3] : e[4*(J)+1]) * inv;                               \
    const float bx_lo = brow<4*(J)+0>(i0, i1, i2, i3);                        \
    const float bx_hi = brow<4*(J)+2>(i0, i1, i2, i3);                        \
    const float by_lo = brow<4*(J)+1>(i0, i1, i2, i3);                        \
    const float by_hi = brow<4*(J)+3>(i0, i1, i2, i3);                        \
    b.x = fmaf(h, bx_hi - bx_lo, bx_lo);                                      \
    b.y = fmaf(h, by_hi - by_lo, by_lo);                                      \
    acc = __builtin_amdgcn_wmma_f32_16x16x4_f32(                              \
        false, a, false, b, (short)0, acc, false, false);                     \
  }
  WMMA_STEP(0)
  WMMA_STEP(1)
  WMMA_STEP(2)
  WMMA_STEP(3)
#undef WMMA_STEP

  if (n < 4) {
    #pragma unroll
    for (int v = 0; v < 8; ++v) {
      const int r = v + (hi ? 8 : 0);      // D row per ISA C/D layout
      coef[(size_t)(obase + r) * 4 + n] = acc[v];
    }
  }
}

// Main memory-bound kernel: async-stage TB contiguous x rows into LDS (CDNA5
// async path, no VGPR round trip), then gather a,b from LDS and emit the fused
// bilinear form with coalesced nontemporal stores.
__global__ __launch_bounds__(256) void logic_kernel(const float* __restrict__ x,
                                                    const int* __restrict__ ca,
                                                    const int* __restrict__ cb,
                                                    const float4* __restrict__ coef,
                                                    float* __restrict__ out) {
  __shared__ __align__(16) float smem[TB * IN_DIM];
  const int tid    = threadIdx.x;
  const int batch0 = blockIdx.x * TB;

  // 64KB contiguous global -> LDS, 16B per lane per issue.
  const uint64_t gbase = (uint64_t)(uintptr_t)(x + (size_t)batch0 * IN_DIM);
  const uint32_t lds0  = (uint32_t)(uintptr_t)(&smem[0]);  // low 32 bits = LDS byte offset
  #pragma unroll 4
  for (int i = tid; i < (TB * IN_DIM) / 4; i += 256) {
    const uint32_t off   = (uint32_t)i * 16u;
    const uint32_t laddr = lds0 + off;
    asm volatile("global_load_async_to_lds_b128 %0, %1, %2"
                 :: "v"(laddr), "v"(off), "s"(gbase) : "memory");
  }
  asm volatile("s_wait_asynccnt 0x0" ::: "memory");
  __syncthreads();

  for (int o = tid; o < OUT_DIM; o += 256) {
    const int    ia = ca[o];
    const int    ib = cb[o];
    const float4 c  = coef[o];
    #pragma unroll
    for (int r = 0; r < TB; ++r) {
      const float a = smem[r * IN_DIM + ia];
      const float b = smem[r * IN_DIM + ib];
      const float v = fmaf(c.w * a, b, fmaf(c.z, b, fmaf(c.y, a, c.x)));
      // write-once output stream: nontemporal keeps L2 free for conn/coef reuse
      __builtin_nontemporal_store(v, &out[(size_t)(batch0 + r) * OUT_DIM + o]);
    }
  }
}

extern "C" void kernel_launch(void* const* d_in, const int* in_sizes, int n_in,
                              void* d_out, int out_size, void* d_ws, size_t ws_size,
                              hipStream_t stream) {
  const float* x  = (const float*)d_in[0];
  const float* w  = (const float*)d_in[1];
  const int*   ca = (const int*)d_in[2];
  const int*   cb = (const int*)d_in[3];
  float*       out  = (float*)d_out;
  float*       coef = (float*)d_ws;   // OUT_DIM*4 floats = 128 KB scratch

  coef_kernel<<<OUT_DIM / 128, 256, 0, stream>>>(w, coef);
  logic_kernel<<<BATCH / TB, 256, 0, stream>>>(x, ca, cb, (const float4*)coef, out);
}